// MorphTEmbedding_18622978196266
// MI455X (gfx1250) — compile-verified
//
#include <hip/hip_runtime.h>

typedef float v2f __attribute__((ext_vector_type(2)));
typedef float v8f __attribute__((ext_vector_type(8)));

#define TOKS_PER_BLOCK 8
#define EMB_DIM 512
#define LN_EPS 1e-5f
// weight layout: (RANK=8, NUM_MORPH=10000, CORE_DIM=8) -> elem(r,m,d) at r*80000 + m*8 + d

__launch_bounds__(256, 2)
__global__ void morpht_emb_kernel(const int* __restrict__ x,
                                  const int* __restrict__ co,
                                  const float* __restrict__ weight,
                                  const float* __restrict__ ln_scale,
                                  const float* __restrict__ ln_bias,
                                  float* __restrict__ out,
                                  int n_tok)
{
    __shared__ __align__(16) float cores[TOKS_PER_BLOCK][192];    // [tok][mat*64 + r*8 + d]
    __shared__ __align__(16) float wstage[TOKS_PER_BLOCK][EMB_DIM];
    __shared__ int midx[TOKS_PER_BLOCK][3];

    const int tid  = threadIdx.x;
    const int tok0 = blockIdx.x * TOKS_PER_BLOCK;

    // Resolve morph ids for the block's 8 tokens
    if (tid < TOKS_PER_BLOCK * 3) {
        int tok = tid / 3, mat = tid - tok * 3;
        int t = tok0 + tok; if (t >= n_tok) t = n_tok - 1;
        midx[tok][mat] = co[x[t] * 3 + mat];
    }
    __syncthreads();

    // Cooperative gather of 8 tokens x 3 cores x 8x8 fp32 (groups of 8 lanes load
    // 32B contiguous rows; weight (2.56MB) stays L2-resident)
    for (int idx = tid; idx < TOKS_PER_BLOCK * 192; idx += 256) {
        int tok = idx / 192;
        int rem = idx - tok * 192;
        int mat = rem >> 6;
        int r   = (rem >> 3) & 7;
        int d   = rem & 7;
        cores[tok][rem] = weight[r * 80000 + midx[tok][mat] * 8 + d];
    }
    __syncthreads();

    const int wave = tid >> 5;      // one token per wave
    const int lane = tid & 31;
    const int h    = lane >> 4;     // lane half (drives K placement per ISA layout)
    const int li   = lane & 15;

    const float* c0 = &cores[wave][0];
    const float* c1 = &cores[wave][64];
    const float* c2 = &cores[wave][128];

    // B fragments (4x16, K x N): N = li, cols >= 8 zero-padded.
    // lanes 0-15: VGPR{0,1} = K{0,1}; lanes 16-31: K{2,3}  (+4 for kstep 1)
    v2f bfrag[2];
    #pragma unroll
    for (int ks = 0; ks < 2; ++ks) {
        int r0 = ks * 4 + 2 * h;
        v2f b;
        b.x = (li < 8) ? c2[r0 * 8 + li]       : 0.0f;
        b.y = (li < 8) ? c2[(r0 + 1) * 8 + li] : 0.0f;
        bfrag[ks] = b;
    }

    // W(64x8) = P^T(64x8) x C(8x8) via 4 M-tiles x 2 K-steps of v_wmma_f32_16x16x4_f32
    // P[r, ab] = core0[r, ab>>3] * core1[r, ab&7]
    v8f acc[4];
    #pragma unroll
    for (int mt = 0; mt < 4; ++mt) {
        int ab = mt * 16 + li;
        int a  = ab >> 3;
        int b  = ab & 7;
        v8f c = {0.f, 0.f, 0.f, 0.f, 0.f, 0.f, 0.f, 0.f};
        #pragma unroll
        for (int ks = 0; ks < 2; ++ks) {
            int r0 = ks * 4 + 2 * h;
            v2f af;
            af.x = c0[r0 * 8 + a] * c1[r0 * 8 + b];
            af.y = c0[(r0 + 1) * 8 + a] * c1[(r0 + 1) * 8 + b];
            c = __builtin_amdgcn_wmma_f32_16x16x4_f32(false, af, false, bfrag[ks],
                                                      (short)0, c, false, false);
        }
        acc[mt] = c;
    }

    // LayerNorm statistics: padded columns are exactly 0, so sum over all 32
    // lanes x 32 regs covers exactly the 512 valid values.
    float s = 0.f, sq = 0.f;
    #pragma unroll
    for (int mt = 0; mt < 4; ++mt)
        #pragma unroll
        for (int p = 0; p < 8; ++p) {
            float v = acc[mt][p];
            s += v; sq += v * v;
        }
    #pragma unroll
    for (int off = 16; off >= 1; off >>= 1) {
        s  += __shfl_xor(s,  off, 32);
        sq += __shfl_xor(sq, off, 32);
    }
    const float mu  = s * (1.0f / 512.0f);
    const float var = sq * (1.0f / 512.0f) - mu * mu;   // E[x^2] - E[x]^2 (ddof=0)
    const float inv = rsqrtf(var + LN_EPS);

    // Stage valid D columns (j<8) to LDS: row = ab, col = c, flat = ab*8+c
    if (li < 8) {
        #pragma unroll
        for (int mt = 0; mt < 4; ++mt)
            #pragma unroll
            for (int p = 0; p < 8; ++p) {
                int row = mt * 16 + p + 8 * h;
                wstage[wave][row * 8 + li] = acc[mt][p];
            }
    }
    __syncthreads();

    // Coalesced 512B-per-burst stores with LN applied on the fly
    int t = tok0 + wave; if (t >= n_tok) t = n_tok - 1;
    float* __restrict__ orow = out + (size_t)t * EMB_DIM;
    #pragma unroll
    for (int k = 0; k < 4; ++k) {
        int idx = k * 128 + lane * 4;
        float4 v  = *(const float4*)&wstage[wave][idx];
        float4 sc = *(const float4*)&ln_scale[idx];
        float4 bi = *(const float4*)&ln_bias[idx];
        float4 o;
        o.x = (v.x - mu) * inv * sc.x + bi.x;
        o.y = (v.y - mu) * inv * sc.y + bi.y;
        o.z = (v.z - mu) * inv * sc.z + bi.z;
        o.w = (v.w - mu) * inv * sc.w + bi.w;
        *(float4*)&orow[idx] = o;
    }
}

extern "C" void kernel_launch(void* const* d_in, const int* in_sizes, int n_in,
                              void* d_out, int out_size, void* d_ws, size_t ws_size,
                              hipStream_t stream) {
    const int*   x       = (const int*)d_in[0];    // (8, 2048) token surface ids
    const int*   co      = (const int*)d_in[1];    // (50000, 3) morph ids
    const float* weight  = (const float*)d_in[2];  // (8, 10000, 8)
    const float* ln_s    = (const float*)d_in[3];  // (512,)
    const float* ln_b    = (const float*)d_in[4];  // (512,)
    float*       out     = (float*)d_out;          // (8, 2048, 512) fp32

    const int n_tok  = in_sizes[0];                // 16384
    const int blocks = (n_tok + TOKS_PER_BLOCK - 1) / TOKS_PER_BLOCK;
    morpht_emb_kernel<<<blocks, 256, 0, stream>>>(x, co, weight, ln_s, ln_b, out, n_tok);
}